// YoloNASOBBAssigner_42073499631799
// MI455X (gfx1250) — compile-verified
//
#include <hip/hip_runtime.h>
#include <hip/hip_bf16.h>
#include <cstdint>
#include <cstddef>

#define B_    32
#define L_    8400
#define C_    80
#define N_    64
#define TOPK_ 13
#define EPSF  1e-9f

typedef float v2f __attribute__((ext_vector_type(2)));
typedef float v8f __attribute__((ext_vector_type(8)));
typedef int   asyncv4 __attribute__((vector_size(16)));   // matches builtin param type

#if __has_builtin(__builtin_amdgcn_global_load_async_to_lds_b128) && \
    __has_builtin(__builtin_amdgcn_s_wait_asynccnt)
#define HAVE_ASYNC_LDS 1
#endif

struct Work {
  float* cls;                                         // [B,N,L] one-hot-GEMM result
  float *pminx, *pmaxx, *pminy, *pmaxy, *parea;       // [B*L] pred AABB
  float *gminx, *gmaxx, *gminy, *gmaxy, *garea;       // [B*N] gt AABB
  float *gcx, *gcy, *gw2, *gh2, *gcs, *gsn, *gpad;    // [B*N] gt params
  int*   cand;                                        // [B*N*13] candidate l (or -1)
  int*   assigned;                                    // [B*L] assigned g (or -1)
  float* aligna;                                      // [B*L] align at assignment
  int*   maxm;                                        // [B*N] f32 bits, per-gt max align
  int*   maxi;                                        // [B*N] f32 bits, per-gt max iou
};

__device__ __forceinline__ float iou_aabb(float ax1, float ax2, float ay1, float ay2, float aa,
                                          float bx1, float bx2, float by1, float by2, float ba) {
  float iw = fmaxf(fminf(ax2, bx2) - fmaxf(ax1, bx1), 0.f);
  float ih = fmaxf(fminf(ay2, by2) - fmaxf(ay1, by1), 0.f);
  float inter = iw * ih;
  float iou = inter / (aa + ba - inter + EPSF);
  return fminf(fmaxf(iou, 0.f), 1.f);
}

// ---------------- K1a: pred rbox -> AABB (SoA) ----------------
__global__ __launch_bounds__(256) void pre_pred(const float* __restrict__ rb, Work w) {
  int i = blockIdx.x * 256 + threadIdx.x;
  if (i >= B_ * L_) return;
  float cx = rb[i*5+0], cy = rb[i*5+1], ww = rb[i*5+2], hh = rb[i*5+3], r = rb[i*5+4];
  float sn, cs; sincosf(r, &sn, &cs);
  float dx = 0.5f*ww*cs, dy = 0.5f*hh*sn;           // reference's (non-standard) corner basis
  float ex = fmaxf(fabsf(dx*cs + dy*sn), fabsf(dx*cs - dy*sn));
  float ey = fmaxf(fabsf(dx*sn + dy*cs), fabsf(dx*sn - dy*cs));
  w.pminx[i] = cx - ex; w.pmaxx[i] = cx + ex;
  w.pminy[i] = cy - ey; w.pmaxy[i] = cy + ey;
  w.parea[i] = ww * hh;
}

// ---------------- K1b: gt rbox -> AABB + params (SoA) ----------------
__global__ __launch_bounds__(256) void pre_gt(const float* __restrict__ rb,
                                              const float* __restrict__ pad, Work w) {
  int i = blockIdx.x * 256 + threadIdx.x;
  if (i >= B_ * N_) return;
  float cx = rb[i*5+0], cy = rb[i*5+1], ww = rb[i*5+2], hh = rb[i*5+3], r = rb[i*5+4];
  float sn, cs; sincosf(r, &sn, &cs);
  float dx = 0.5f*ww*cs, dy = 0.5f*hh*sn;
  float ex = fmaxf(fabsf(dx*cs + dy*sn), fabsf(dx*cs - dy*sn));
  float ey = fmaxf(fabsf(dx*sn + dy*cs), fabsf(dx*sn - dy*cs));
  w.gminx[i] = cx - ex; w.gmaxx[i] = cx + ex;
  w.gminy[i] = cy - ey; w.gmaxy[i] = cy + ey;
  w.garea[i] = ww * hh;
  w.gcx[i] = cx; w.gcy[i] = cy; w.gw2[i] = 0.5f*ww; w.gh2[i] = 0.5f*hh;
  w.gcs[i] = cs; w.gsn[i] = sn; w.gpad[i] = pad[i];
}

// ---------------- K2: one-hot GEMM via V_WMMA_F32_16X16X4_F32 ----------------
// cls[b,g,l] = sum_c onehot(label[b,g])[c] * pred_scores[b,l,c]  (exact in f32)
// PPITCH=84: rows stay 16B-aligned (84*4=336) for b128 LDS fills, and the
// 16-lane column read stride (20 banks mod 64, 64/gcd(20,64)=16 period) is
// bank-conflict-free; half-wave +2 column offset cannot collide (20k==2 mod 64
// has no solution).
#define PPITCH 84
__global__ __launch_bounds__(256) void cls_wmma(const float* __restrict__ ps,
                                                const int* __restrict__ labels, Work w) {
  __shared__ float Pt[128 * PPITCH];
  __shared__ int lab[N_];
  const int b   = blockIdx.y;
  const int l0  = blockIdx.x * 128;
  const int tid = threadIdx.x;
  if (tid < N_) lab[tid] = labels[b * N_ + tid];
  // stage 128 x 80 f32 tile in 16B chunks: 128 rows * 20 chunks
  for (int ch = tid; ch < 128 * 20; ch += 256) {
    const int row = ch / 20, c4 = (ch % 20) * 4;
    const int l = l0 + row;
    if (l < L_) {
#ifdef HAVE_ASYNC_LDS
      __builtin_amdgcn_global_load_async_to_lds_b128(
          (__attribute__((address_space(1))) asyncv4*)(ps + ((size_t)b * L_ + l) * C_ + c4),
          (__attribute__((address_space(3))) asyncv4*)(&Pt[row * PPITCH + c4]),
          0, 0);
#else
      *(float4*)&Pt[row * PPITCH + c4] =
          *(const float4*)(ps + ((size_t)b * L_ + l) * C_ + c4);
#endif
    } else {
      float4 z; z.x = 0.f; z.y = 0.f; z.z = 0.f; z.w = 0.f;
      *(float4*)&Pt[row * PPITCH + c4] = z;
    }
  }
#ifdef HAVE_ASYNC_LDS
  __builtin_amdgcn_s_wait_asynccnt(0);   // barrier does not cover ASYNCcnt
#endif
  __syncthreads();
  const int lane = tid & 31;
  const int wave = tid >> 5;
  const int lr   = lane & 15;        // M (A) / N (B,D) position
  const int hi   = lane >> 4;        // lanes 16-31 hold K+2 (A,B) / M+8 (D)
  const int kb   = hi * 2;
  const int lrow = wave * 16 + lr;   // l within 128-tile
  v8f a0 = {0}, a1 = {0}, a2 = {0}, a3 = {0};
  for (int k = 0; k < 20; ++k) {     // K = 80 classes, 4 per step
    const int c0 = k * 4 + kb;
    v2f bf; bf.x = Pt[lrow * PPITCH + c0]; bf.y = Pt[lrow * PPITCH + c0 + 1];
    {
      int lb = lab[lr];
      v2f af; af.x = (lb == c0) ? 1.f : 0.f; af.y = (lb == c0 + 1) ? 1.f : 0.f;
      a0 = __builtin_amdgcn_wmma_f32_16x16x4_f32(false, af, false, bf, (short)0, a0, false, false);
    }
    {
      int lb = lab[16 + lr];
      v2f af; af.x = (lb == c0) ? 1.f : 0.f; af.y = (lb == c0 + 1) ? 1.f : 0.f;
      a1 = __builtin_amdgcn_wmma_f32_16x16x4_f32(false, af, false, bf, (short)0, a1, false, false);
    }
    {
      int lb = lab[32 + lr];
      v2f af; af.x = (lb == c0) ? 1.f : 0.f; af.y = (lb == c0 + 1) ? 1.f : 0.f;
      a2 = __builtin_amdgcn_wmma_f32_16x16x4_f32(false, af, false, bf, (short)0, a2, false, false);
    }
    {
      int lb = lab[48 + lr];
      v2f af; af.x = (lb == c0) ? 1.f : 0.f; af.y = (lb == c0 + 1) ? 1.f : 0.f;
      a3 = __builtin_amdgcn_wmma_f32_16x16x4_f32(false, af, false, bf, (short)0, a3, false, false);
    }
  }
  const int l_out = l0 + lrow;
  if (l_out < L_) {
    size_t base = (size_t)b * N_ * L_ + l_out;
    const int go = hi * 8;           // D layout: VGPR i -> M=i (lanes 0-15), M=i+8 (16-31)
    #pragma unroll
    for (int i = 0; i < 8; ++i) {
      w.cls[base + (size_t)( 0 + go + i) * L_] = a0[i];
      w.cls[base + (size_t)(16 + go + i) * L_] = a1[i];
      w.cls[base + (size_t)(32 + go + i) * L_] = a2[i];
      w.cls[base + (size_t)(48 + go + i) * L_] = a3[i];
    }
  }
}

// ---------------- K3: per-(b,g) metric row + exact top-13 in LDS ----------------
__global__ __launch_bounds__(256) void topk_sel(const float* __restrict__ anchors, Work w) {
  __shared__ float met[L_];
  __shared__ float rv[256];
  __shared__ int   ri[256];
  __shared__ int   sel[TOPK_];
  const int b = blockIdx.y, g = blockIdx.x, tid = threadIdx.x;
  const int gi = b * N_ + g;
  const float gx1 = w.gminx[gi], gx2 = w.gmaxx[gi], gy1 = w.gminy[gi], gy2 = w.gmaxy[gi];
  const float ga  = w.garea[gi];
  const float gcx = w.gcx[gi], gcy = w.gcy[gi], gw2 = w.gw2[gi], gh2 = w.gh2[gi];
  const float gcs = w.gcs[gi], gsn = w.gsn[gi], gpad = w.gpad[gi];
  const float* clsrow = w.cls + (size_t)gi * L_;
  const size_t pb = (size_t)b * L_;
  for (int l = tid; l < L_; l += 256) {
    float c   = clsrow[l];
    float iou = iou_aabb(gx1, gx2, gy1, gy2, ga,
                         w.pminx[pb + l], w.pmaxx[pb + l], w.pminy[pb + l], w.pmaxy[pb + l],
                         w.parea[pb + l]);
    float ax = anchors[2*l], ay = anchors[2*l+1];
    float dx = ax - gcx, dy = ay - gcy;
    float xl = dx*gcs + dy*gsn, yl = -dx*gsn + dy*gcs;
    bool inside = (fabsf(xl) < gw2) && (fabsf(yl) < gh2);
    float i2 = iou * iou;
    met[l] = inside ? c * i2 * i2 * i2 : 0.f;      // align = cls^1 * iou^6, masked
  }
  __syncthreads();
  for (int rnd = 0; rnd < TOPK_; ++rnd) {
    float bv = -1.f; int bi = 0x7fffffff;
    for (int l = tid; l < L_; l += 256) {
      float v = met[l];
      if (v > bv) { bv = v; bi = l; }
    }
    rv[tid] = bv; ri[tid] = bi;
    __syncthreads();
    for (int s = 128; s > 0; s >>= 1) {
      if (tid < s) {
        if (rv[tid + s] > rv[tid] || (rv[tid + s] == rv[tid] && ri[tid + s] < ri[tid])) {
          rv[tid] = rv[tid + s]; ri[tid] = ri[tid + s];
        }
      }
      __syncthreads();
    }
    if (tid == 0) { sel[rnd] = ri[0]; met[ri[0]] = -1.f; }
    __syncthreads();
  }
  if (tid < TOPK_) {
    int l = sel[tid];
    float ax = anchors[2*l], ay = anchors[2*l+1];
    float dx = ax - gcx, dy = ay - gcy;
    float xl = dx*gcs + dy*gsn, yl = -dx*gsn + dy*gcs;
    bool inside = (fabsf(xl) < gw2) && (fabsf(yl) < gh2);
    w.cand[gi * TOPK_ + tid] = (inside && gpad > 0.f) ? l : -1;   // mask *= in_gts * pad
  }
}

// ---------------- zero init ----------------
__global__ __launch_bounds__(256) void zero_i32(int* a, int nelem) {
  int i = blockIdx.x * 256 + threadIdx.x;
  if (i < nelem) a[i] = 0;
}

// ---------------- K4: resolve multi-assignment, per-gt maxes ----------------
__global__ __launch_bounds__(256) void assign_resolve(Work w) {
  __shared__ int   candL[N_ * TOPK_];
  __shared__ float sx1[N_], sx2[N_], sy1[N_], sy2[N_], sa[N_];
  const int b = blockIdx.y, tid = threadIdx.x;
  for (int i = tid; i < N_ * TOPK_; i += 256) candL[i] = w.cand[(size_t)b * N_ * TOPK_ + i];
  if (tid < N_) {
    int gi = b * N_ + tid;
    sx1[tid] = w.gminx[gi]; sx2[tid] = w.gmaxx[gi];
    sy1[tid] = w.gminy[gi]; sy2[tid] = w.gmaxy[gi];
    sa[tid]  = w.garea[gi];
  }
  __syncthreads();
  const int l = blockIdx.x * 256 + tid;
  if (l >= L_) return;
  int mps = 0, firstg = -1;
  for (int g = 0; g < N_; ++g) {
    bool m = false;
    #pragma unroll
    for (int k = 0; k < TOPK_; ++k) m |= (candL[g * TOPK_ + k] == l);
    if (m) { ++mps; if (firstg < 0) firstg = g; }
  }
  const size_t pi = (size_t)b * L_ + l;
  const float px1 = w.pminx[pi], px2 = w.pmaxx[pi], py1 = w.pminy[pi], py2 = w.pmaxy[pi];
  const float pa  = w.parea[pi];
  int gA = -1;
  if (mps == 1) gA = firstg;
  else if (mps > 1) {                       // replace column with one-hot(argmax_g iou)
    float best = -1.f; int bg = 0;
    for (int g = 0; g < N_; ++g) {
      float iou = iou_aabb(sx1[g], sx2[g], sy1[g], sy2[g], sa[g], px1, px2, py1, py2, pa);
      if (iou > best) { best = iou; bg = g; }   // ties -> lowest g (jnp.argmax)
    }
    gA = bg;
  }
  w.assigned[pi] = gA;
  float alignv = 0.f;
  if (gA >= 0) {
    float iou = iou_aabb(sx1[gA], sx2[gA], sy1[gA], sy2[gA], sa[gA], px1, px2, py1, py2, pa);
    float c = w.cls[((size_t)(b * N_ + gA)) * L_ + l];
    float i2 = iou * iou;
    alignv = c * i2 * i2 * i2;
    atomicMax(&w.maxm[b * N_ + gA], __float_as_int(alignv));   // values >= 0: int order == float order
    atomicMax(&w.maxi[b * N_ + gA], __float_as_int(iou));
  }
  w.aligna[pi] = alignv;
}

// ---------------- K5: emit (labels, rboxes, scores, agi, crowd) ----------------
__global__ __launch_bounds__(256) void finalize(const int* __restrict__ gt_labels,
                                                const float* __restrict__ gt_bboxes,
                                                const int* __restrict__ gt_crowd,
                                                const int* __restrict__ bgp,
                                                Work w, float* __restrict__ out) {
  const int b = blockIdx.y;
  const int l = blockIdx.x * 256 + threadIdx.x;
  if (l >= L_) return;
  const int bg = bgp[0];
  const size_t p  = (size_t)b * L_ + l;
  const size_t BL = (size_t)B_ * L_;
  const int gA = w.assigned[p];
  const bool valid = (gA >= 0);
  const int g  = valid ? gA : 0;            // argmax of all-zero column -> 0
  const int gi = b * N_ + g;
  const int lbl = valid ? gt_labels[gi] : bg;
  float am = 0.f;
  if (valid) {
    float mm = __int_as_float(w.maxm[gi]);
    float mi = __int_as_float(w.maxi[gi]);
    am = w.aligna[p] / (mm + EPSF) * mi;
  }
  const int crowd = gt_crowd[gi];
  out[p] = (float)lbl;                                       // labels
  #pragma unroll
  for (int j = 0; j < 5; ++j)                                // rboxes (gathered even if bg)
    out[BL + p * 5 + j] = gt_bboxes[(size_t)gi * 5 + j];
  const float s = (crowd == 0) ? am : 0.f;                   // scores: one-hot * am * !crowd
  float4* srow = (float4*)(out + BL * 6 + p * C_);
  #pragma unroll
  for (int q = 0; q < C_ / 4; ++q) {
    float4 v; int c = q * 4;
    v.x = (c     == lbl) ? s : 0.f;
    v.y = (c + 1 == lbl) ? s : 0.f;
    v.z = (c + 2 == lbl) ? s : 0.f;
    v.w = (c + 3 == lbl) ? s : 0.f;
    srow[q] = v;
  }
  out[BL * 86 + p] = (float)(b * N_ + g);                    // agi
  out[BL * 87 + p] = (float)crowd;                           // crowd
}

extern "C" void kernel_launch(void* const* d_in, const int* in_sizes, int n_in,
                              void* d_out, int out_size, void* d_ws, size_t ws_size,
                              hipStream_t stream) {
  const float* pred_scores = (const float*)d_in[0];
  const float* pred_rboxes = (const float*)d_in[1];
  const float* anchors     = (const float*)d_in[2];
  const int*   gt_labels   = (const int*)d_in[3];
  const float* gt_bboxes   = (const float*)d_in[4];
  // d_in[5] = gt_poses (unused by reference _assign)
  const int*   gt_crowd    = (const int*)d_in[6];
  const float* pad_gt      = (const float*)d_in[7];
  const int*   bgp         = (const int*)d_in[8];
  float* out = (float*)d_out;

  char* base = (char*)d_ws;
  size_t off = 0;
  auto alloc = [&](size_t elems) -> void* { void* p = base + off; off += elems * 4; return p; };

  Work w;
  w.cls   = (float*)alloc((size_t)B_ * N_ * L_);
  w.pminx = (float*)alloc((size_t)B_ * L_);
  w.pmaxx = (float*)alloc((size_t)B_ * L_);
  w.pminy = (float*)alloc((size_t)B_ * L_);
  w.pmaxy = (float*)alloc((size_t)B_ * L_);
  w.parea = (float*)alloc((size_t)B_ * L_);
  w.gminx = (float*)alloc(B_ * N_);
  w.gmaxx = (float*)alloc(B_ * N_);
  w.gminy = (float*)alloc(B_ * N_);
  w.gmaxy = (float*)alloc(B_ * N_);
  w.garea = (float*)alloc(B_ * N_);
  w.gcx   = (float*)alloc(B_ * N_);
  w.gcy   = (float*)alloc(B_ * N_);
  w.gw2   = (float*)alloc(B_ * N_);
  w.gh2   = (float*)alloc(B_ * N_);
  w.gcs   = (float*)alloc(B_ * N_);
  w.gsn   = (float*)alloc(B_ * N_);
  w.gpad  = (float*)alloc(B_ * N_);
  w.cand     = (int*)alloc((size_t)B_ * N_ * TOPK_);
  w.assigned = (int*)alloc((size_t)B_ * L_);
  w.aligna   = (float*)alloc((size_t)B_ * L_);
  w.maxm     = (int*)alloc(B_ * N_);       // maxm/maxi contiguous -> one zero pass
  w.maxi     = (int*)alloc(B_ * N_);

  pre_pred<<<dim3((B_ * L_ + 255) / 256), dim3(256), 0, stream>>>(pred_rboxes, w);
  pre_gt  <<<dim3((B_ * N_ + 255) / 256), dim3(256), 0, stream>>>(gt_bboxes, pad_gt, w);
  cls_wmma<<<dim3((L_ + 127) / 128, B_), dim3(256), 0, stream>>>(pred_scores, gt_labels, w);
  zero_i32<<<dim3((2 * B_ * N_ + 255) / 256), dim3(256), 0, stream>>>(w.maxm, 2 * B_ * N_);
  topk_sel<<<dim3(N_, B_), dim3(256), 0, stream>>>(anchors, w);
  assign_resolve<<<dim3((L_ + 255) / 256, B_), dim3(256), 0, stream>>>(w);
  finalize<<<dim3((L_ + 255) / 256, B_), dim3(256), 0, stream>>>(gt_labels, gt_bboxes, gt_crowd,
                                                                 bgp, w, out);
}